// ConditionalFeedForward_88553635709706
// MI455X (gfx1250) — compile-verified
//
#include <hip/hip_runtime.h>
#include <hip/hip_bf16.h>

#define NUM_EXPERTS 8
#define HIDDEN 1024
#define INTER 2816
#define NPAIR 32   // T(16) * TOP_K(2)

#if defined(__HIP_DEVICE_COMPILE__) && !__has_builtin(__builtin_amdgcn_wmma_f32_16x16x4_f32)
#error "wmma_f32_16x16x4_f32 builtin not available on this toolchain (device pass)"
#endif

typedef float v2f __attribute__((ext_vector_type(2)));
typedef float v8f __attribute__((ext_vector_type(8)));

// Non-temporal float2 load for single-use weight streams (TH=NT).
__device__ __forceinline__ v2f ldw(const float* p) {
    return __builtin_nontemporal_load((const v2f*)p);
}
__device__ __forceinline__ v2f ldc(const float* p) {
    return *(const v2f*)p;
}

// ---------------------------------------------------------------------------
// Kernel 0: routing. Build per-expert lists of pair ids (pair = t*TOP_K + a).
// ---------------------------------------------------------------------------
__global__ void moe_route(const int* __restrict__ idx,
                          int* __restrict__ counts,
                          int* __restrict__ lists) {
    int e = threadIdx.x;
    if (e >= NUM_EXPERTS) return;
    int c = 0;
    for (int p = 0; p < NPAIR; ++p) {
        if (idx[p] == e) lists[e * NPAIR + c++] = p;
    }
    counts[e] = c;
}

// ---------------------------------------------------------------------------
// Kernel 1: grouped GEMM for gate & "down" branches + SiLU fuse.
//   h[pair][o] = silu(x[t]·gate[e][o]) * (x[t]·down[e][o])
// One wave owns a 16-wide o-tile for one expert; K loop streams HIDDEN=1024.
// WMMA mapping (f32 16x16x4):
//   lane L: n = L&15 (N / B-row o, D column), khalf = (L>>4)*2
//   A v2f = x[pair_row(n)][k0+khalf .. +1]   (rows = pairs)
//   B v2f = w[o0+n][k0+khalf .. +1]          (w row-major [INTER][HIDDEN])
//   D v8f: lane holds D[M = r + 8*(L>>4)][N = n]
// ---------------------------------------------------------------------------
__global__ __launch_bounds__(256) void moe_gate_up(
    const float* __restrict__ x,
    const float* __restrict__ gate,
    const float* __restrict__ down,
    const int* __restrict__ counts,
    const int* __restrict__ lists,
    float* __restrict__ h)
{
    const int e = blockIdx.x;
    const int M = counts[e];
    if (M == 0) return;                       // uniform: never touch weights

    const int wave  = threadIdx.x >> 5;
    const int lane  = threadIdx.x & 31;
    const int nn    = lane & 15;
    const int khalf = (lane >> 4) << 1;       // 0 or 2
    const int n0    = (blockIdx.y * 8 + wave) * 16;

    const float* gb = gate + (size_t)e * INTER * HIDDEN + (size_t)(n0 + nn) * HIDDEN + khalf;
    const float* db = down + (size_t)e * INTER * HIDDEN + (size_t)(n0 + nn) * HIDDEN + khalf;

    const int mtiles = (M + 15) >> 4;
    for (int mt = 0; mt < mtiles; ++mt) {
        // A-row for this lane (both half-waves with same nn pick the same row).
        const int mA   = mt * 16 + nn;
        const int pA   = (mA < M) ? lists[e * NPAIR + mA] : 0;  // pad -> row 0 (discarded)
        const float* xa = x + (size_t)(pA >> 1) * HIDDEN + khalf;

        v8f accg = {};
        v8f accd = {};
        #pragma unroll 4
        for (int k0 = 0; k0 < HIDDEN; k0 += 4) {
            v2f av = ldc(xa + k0);
            v2f gv = ldw(gb + k0);
            v2f dv = ldw(db + k0);
            accg = __builtin_amdgcn_wmma_f32_16x16x4_f32(false, av, false, gv,
                                                         (short)0, accg, false, false);
            accd = __builtin_amdgcn_wmma_f32_16x16x4_f32(false, av, false, dv,
                                                         (short)0, accd, false, false);
        }

        const int mhi = (lane >> 4) * 8;
        #pragma unroll
        for (int r = 0; r < 8; ++r) {
            const int mrow = mt * 16 + mhi + r;
            if (mrow < M) {
                const int pr = lists[e * NPAIR + mrow];
                const float g = accg[r];
                const float d = accd[r];
                const float s = g / (1.0f + __expf(-g));   // silu(g)
                h[(size_t)pr * INTER + n0 + nn] = s * d;
            }
        }
    }
}

// ---------------------------------------------------------------------------
// Kernel 2: grouped GEMM for the final contraction over INTER=2816.
//   out[pair][i] = sum_o h[pair][o] * up[e][i][o]   (up row-major [HIDDEN][INTER])
// ---------------------------------------------------------------------------
__global__ __launch_bounds__(256) void moe_down(
    const float* __restrict__ h,
    const float* __restrict__ up,
    const int* __restrict__ counts,
    const int* __restrict__ lists,
    float* __restrict__ out)
{
    const int e = blockIdx.x;
    const int M = counts[e];
    if (M == 0) return;

    const int wave  = threadIdx.x >> 5;
    const int lane  = threadIdx.x & 31;
    const int nn    = lane & 15;
    const int khalf = (lane >> 4) << 1;
    const int i0    = (blockIdx.y * 8 + wave) * 16;

    const float* ub = up + (size_t)e * HIDDEN * INTER + (size_t)(i0 + nn) * INTER + khalf;

    const int mtiles = (M + 15) >> 4;
    for (int mt = 0; mt < mtiles; ++mt) {
        const int mA = mt * 16 + nn;
        const int pA = (mA < M) ? lists[e * NPAIR + mA] : 0;
        const float* ha = h + (size_t)pA * INTER + khalf;

        v8f acc = {};
        #pragma unroll 4
        for (int k0 = 0; k0 < INTER; k0 += 4) {
            v2f av = ldc(ha + k0);
            v2f bv = ldw(ub + k0);
            acc = __builtin_amdgcn_wmma_f32_16x16x4_f32(false, av, false, bv,
                                                        (short)0, acc, false, false);
        }

        const int mhi = (lane >> 4) * 8;
        #pragma unroll
        for (int r = 0; r < 8; ++r) {
            const int mrow = mt * 16 + mhi + r;
            if (mrow < M) {
                const int pr = lists[e * NPAIR + mrow];
                out[(size_t)pr * HIDDEN + i0 + nn] = acc[r];
            }
        }
    }
}

// ---------------------------------------------------------------------------
// Launch. Inputs (setup_inputs order):
//   0: x            [16,1024]      f32
//   1: expert_idx   [16,2]         i32
//   2: gate_proj    [8,2816,1024]  f32
//   3: up_proj      [8,1024,2816]  f32
//   4: down_proj    [8,2816,1024]  f32
// Output: [16,2,1024] f32
// ---------------------------------------------------------------------------
extern "C" void kernel_launch(void* const* d_in, const int* in_sizes, int n_in,
                              void* d_out, int out_size, void* d_ws, size_t ws_size,
                              hipStream_t stream) {
    const float* x    = (const float*)d_in[0];
    const int*   idx  = (const int*)  d_in[1];
    const float* gate = (const float*)d_in[2];
    const float* up   = (const float*)d_in[3];
    const float* down = (const float*)d_in[4];
    float*       out  = (float*)d_out;

    // Workspace layout: counts[8] | lists[8][32] | (pad to 2KB) | h[32][2816] f32
    int*   counts = (int*)d_ws;
    int*   lists  = counts + NUM_EXPERTS;
    float* h      = (float*)((char*)d_ws + 2048);

    moe_route<<<dim3(1), dim3(32), 0, stream>>>(idx, counts, lists);
    // INTER tiles: 2816 / (8 waves * 16) = 22 blocks per expert
    moe_gate_up<<<dim3(NUM_EXPERTS, 22), dim3(256), 0, stream>>>(x, gate, down, counts, lists, h);
    // HIDDEN tiles: 1024 / (8 waves * 16) = 8 blocks per expert
    moe_down<<<dim3(NUM_EXPERTS, 8), dim3(256), 0, stream>>>(h, up, counts, lists, out);
}